// AccumulatorRaceDecisionV2_75943611728369
// MI455X (gfx1250) — compile-verified
//
#include <hip/hip_runtime.h>
#include <stdint.h>

#define T_STEPS 120

typedef int v4i __attribute__((ext_vector_type(4)));

#if __has_builtin(__builtin_amdgcn_global_load_async_to_lds_b128) && \
    __has_builtin(__builtin_amdgcn_s_wait_asynccnt)
#define USE_ASYNC_LDS 1
#endif

__device__ __forceinline__ uint32_t rotl32(uint32_t x, int r) {
  return (x << r) | (x >> (32 - r));
}

struct TFOut { uint32_t x, y; };

// Threefry-2x32, 20 rounds — exactly JAX's threefry2x32. Both output lanes used.
__device__ __forceinline__ TFOut threefry2x32(uint32_t k0, uint32_t k1,
                                              uint32_t x0, uint32_t x1) {
  uint32_t k2 = k0 ^ k1 ^ 0x1BD11BDAu;
  x0 += k0; x1 += k1;
#define TF_R(r) { x0 += x1; x1 = rotl32(x1, r); x1 ^= x0; }
  TF_R(13) TF_R(15) TF_R(26) TF_R(6)
  x0 += k1; x1 += k2 + 1u;
  TF_R(17) TF_R(29) TF_R(16) TF_R(24)
  x0 += k2; x1 += k0 + 2u;
  TF_R(13) TF_R(15) TF_R(26) TF_R(6)
  x0 += k0; x1 += k1 + 3u;
  TF_R(17) TF_R(29) TF_R(16) TF_R(24)
  x0 += k1; x1 += k2 + 4u;
  TF_R(13) TF_R(15) TF_R(26) TF_R(6)
  x0 += k2; x1 += k0 + 5u;
#undef TF_R
  TFOut o; o.x = x0; o.y = x1; return o;
}

__device__ __forceinline__ float softplus_f(float x) {
  // jax.nn.softplus == max(x,0) + log1p(exp(-|x|))
  return fmaxf(x, 0.0f) + log1pf(__expf(-fabsf(x)));
}

// Single-precision erfinv (Giles 2010), valid for |x| <= 1 - 6e-8.
__device__ __forceinline__ float erfinv_f(float x) {
  float w = -__logf((1.0f - x) * (1.0f + x));
  float p;
  if (w < 5.0f) {
    w -= 2.5f;
    p =              2.81022636e-08f;
    p = fmaf(p, w,   3.43273939e-07f);
    p = fmaf(p, w,  -3.5233877e-06f);
    p = fmaf(p, w,  -4.39150654e-06f);
    p = fmaf(p, w,   0.00021858087f);
    p = fmaf(p, w,  -0.00125372503f);
    p = fmaf(p, w,  -0.00417768164f);
    p = fmaf(p, w,   0.246640727f);
    p = fmaf(p, w,   1.50140941f);
  } else {
    w = sqrtf(w) - 3.0f;
    p =             -0.000200214257f;
    p = fmaf(p, w,   0.000100950558f);
    p = fmaf(p, w,   0.00134934322f);
    p = fmaf(p, w,  -0.00367342844f);
    p = fmaf(p, w,   0.00573950773f);
    p = fmaf(p, w,  -0.0076224613f);
    p = fmaf(p, w,   0.00943887047f);
    p = fmaf(p, w,   1.00167406f);
    p = fmaf(p, w,   2.83297682f);
  }
  return p * x;
}

// JAX normal: uniform in [nextafter(-1,0), 1), then sqrt(2)*erfinv(u).
__device__ __forceinline__ float bits_to_normal(uint32_t bits) {
  const float lo = -0.99999994f; // nextafter(-1, 0) in f32
  float f = __uint_as_float((bits >> 9) | 0x3F800000u) - 1.0f; // [0,1)
  float u = fmaf(f, 1.0f - lo, lo);
  u = fmaxf(u, lo);
  return 1.41421356237f * erfinv_f(u);
}

// One thread owns the ROW PAIR (i, i + B/2): element e = 4i+c (low counter
// half) and e+half (high counter half) come from the SAME threefry block as
// lane0/lane1, so both 32-bit outputs are consumed and total Threefry work
// is halved vs one-row-per-thread.
__global__ __launch_bounds__(256)
void acc_race_kernel(const float* __restrict__ logits,
                     const float* __restrict__ input_scale,
                     const float* __restrict__ leak,
                     const float* __restrict__ self_exc,
                     const float* __restrict__ inhibition,
                     const float* __restrict__ noise_std,
                     const float* __restrict__ evidence_w,
                     const float* __restrict__ evidence_b,
                     float* __restrict__ out, int B) {
  __shared__ uint32_t sKeys[2 * T_STEPS];
  __shared__ float    sLog[256 * 8];   // 2 rows x 4 classes per thread

  const int tid   = threadIdx.x;
  const int halfB = B >> 1;
  const int i     = blockIdx.x * 256 + tid;   // row-pair index in [0, B/2)
  const bool active = (i < halfB);

  // ---- Stage both rows' logits into LDS (CDNA5 async-LDS path) ----
  if (active) {
#ifdef USE_ASYNC_LDS
    __builtin_amdgcn_global_load_async_to_lds_b128(
        (v4i*)(logits + (size_t)i * 4),
        (v4i*)(&sLog[tid * 8]), 0, 0);
    __builtin_amdgcn_global_load_async_to_lds_b128(
        (v4i*)(logits + (size_t)(i + halfB) * 4),
        (v4i*)(&sLog[tid * 8 + 4]), 0, 0);
#else
    *(float4*)&sLog[tid * 8]     = *(const float4*)(logits + (size_t)i * 4);
    *(float4*)&sLog[tid * 8 + 4] = *(const float4*)(logits + (size_t)(i + halfB) * 4);
#endif
  }

  // ---- Per-step subkeys, exactly jax.random.split(key(42), 120) ----
  // flat = concat(lane0 of blocks (j, j+120) for j<120,
  //               lane1 of blocks (j-120, j) for j>=120); subkey t = flat[2t],flat[2t+1]
  const uint32_t K0 = 0u, K1 = 42u; // jax.random.key(42)
  if (tid < T_STEPS) {
    uint32_t w0, w1;
    uint32_t j0 = 2u * (uint32_t)tid, j1 = j0 + 1u;
    if (tid < T_STEPS / 2) {
      w0 = threefry2x32(K0, K1, j0, j0 + T_STEPS).x;
      w1 = threefry2x32(K0, K1, j1, j1 + T_STEPS).x;
    } else {
      w0 = threefry2x32(K0, K1, j0 - T_STEPS, j0).y;
      w1 = threefry2x32(K0, K1, j1 - T_STEPS, j1).y;
    }
    sKeys[2 * tid]     = w0;
    sKeys[2 * tid + 1] = w1;
  }
#ifdef USE_ASYNC_LDS
  __builtin_amdgcn_s_wait_asynccnt(0);
#endif
  __syncthreads();

  if (!active) return;

  // ---- Evidence + softplus'd parameters (8 channels: [row0 c0..3, row1 c0..3]) ----
  const float scale = input_scale[0];
  const float ew = evidence_w[0], ebv = evidence_b[0];
  float ev[8], lk[4], se[4];
#pragma unroll
  for (int r = 0; r < 2; ++r) {
    float lg[4];
#pragma unroll
    for (int c = 0; c < 4; ++c) lg[c] = sLog[tid * 8 + r * 4 + c] * scale;
    const float mean = 0.25f * (lg[0] + lg[1] + lg[2] + lg[3]);
#pragma unroll
    for (int c = 0; c < 4; ++c)
      ev[r * 4 + c] = softplus_f(lg[c] - mean) * ew + ebv;
  }
#pragma unroll
  for (int c = 0; c < 4; ++c) {
    lk[c] = softplus_f(leak[c]);
    se[c] = softplus_f(self_exc[c]);
  }
  const float inh = softplus_f(inhibition[0]);
  const float ns  = softplus_f(noise_std[0]);
  // competition_gain unused: MIX == 0.0 exactly.

  // ---- Rollout state ----
  float a[8], diffc[8], slope[8];
  int   idx[8];
#pragma unroll
  for (int q = 0; q < 8; ++q) {
    a[q] = 0.f; diffc[q] = 0.f; slope[q] = 0.f; idx[q] = T_STEPS - 1;
  }
  uint32_t crossed = 0u;

  const uint32_t half = (uint32_t)B * 2u;   // (B*C)/2 counter split point
  const uint32_t e0   = (uint32_t)i * 4u;   // counters for row i (low half)

  for (int t = 0; t < T_STEPS; ++t) {
    const uint32_t k0 = sKeys[2 * t], k1 = sKeys[2 * t + 1];
    const float totLo = a[0] + a[1] + a[2] + a[3];
    const float totHi = a[4] + a[5] + a[6] + a[7];
#pragma unroll
    for (int c = 0; c < 4; ++c) {
      const uint32_t e = e0 + (uint32_t)c;
      const TFOut rb = threefry2x32(k0, k1, e, e + half);
      // lane0 -> (row i, class c); lane1 -> (row i+B/2, class c)
#pragma unroll
      for (int r = 0; r < 2; ++r) {
        const int q = r * 4 + c;
        const float tot  = r ? totHi : totLo;
        const uint32_t bits = r ? rb.y : rb.x;
        const float z = bits_to_normal(bits) * ns;
        const float drive =
            ev[q] + se[c] * a[q] - inh * (tot - a[q]) - lk[c] * a[q];
        const float ds = softplus_f(drive + z) - a[q];
        const float an = fmaxf(fmaf(0.2f, ds, a[q]), 0.0f);
        const float d  = an - 0.5f;
        if (!(crossed & (1u << q)) && d > 0.0f) {
          crossed |= (1u << q);
          idx[q]   = t;
          diffc[q] = d;
          slope[q] = 0.2f * ds;
        }
        a[q] = an;
      }
    }
  }

  // ---- Decision times (seconds) ----
  float oc[8];
#pragma unroll
  for (int q = 0; q < 8; ++q) {
    if (crossed & (1u << q)) {
      const float s  = slope[q];
      const float ss = (s >= 0.0f) ? fmaxf(s, 1e-3f) : fminf(s, -1e-3f);
      const float tc = (float)idx[q] - diffc[q] / ss;
      oc[q] = tc * 0.01f;           // * DT(10 ms) / 1000
    } else {
      oc[q] = 1.2f;                 // T_STEPS * DT / 1000
    }
  }
  float4 o0, o1;
  o0.x = oc[0]; o0.y = oc[1]; o0.z = oc[2]; o0.w = oc[3];
  o1.x = oc[4]; o1.y = oc[5]; o1.z = oc[6]; o1.w = oc[7];
  *(float4*)(out + (size_t)i * 4)             = o0;
  *(float4*)(out + (size_t)(i + halfB) * 4)   = o1;
}

extern "C" void kernel_launch(void* const* d_in, const int* in_sizes, int n_in,
                              void* d_out, int out_size, void* d_ws, size_t ws_size,
                              hipStream_t stream) {
  (void)n_in; (void)d_ws; (void)ws_size; (void)out_size;
  const float* logits     = (const float*)d_in[0];
  const float* in_scale   = (const float*)d_in[1];
  const float* leak       = (const float*)d_in[2];
  const float* self_exc   = (const float*)d_in[3];
  const float* inhibition = (const float*)d_in[4];
  // d_in[5] = competition_gain (unused: MIX == 0)
  const float* noise_std  = (const float*)d_in[6];
  const float* evid_w     = (const float*)d_in[7];
  const float* evid_b     = (const float*)d_in[8];
  float* out = (float*)d_out;

  const int B = in_sizes[0] / 4;     // 65536 rows, C = 4
  const int halfB = B >> 1;          // one thread per row pair
  dim3 block(256);
  dim3 grid((halfB + 255) / 256);
  acc_race_kernel<<<grid, block, 0, stream>>>(
      logits, in_scale, leak, self_exc, inhibition, noise_std,
      evid_w, evid_b, out, B);
}